// ThinPlateSpline2D_22299470200951
// MI455X (gfx1250) — compile-verified
//
#include <hip/hip_runtime.h>
#include <hip/hip_bf16.h>

// ---------------------------------------------------------------------------
// Thin-plate-spline 2D warp, MI455X (gfx1250, wave32).
// Memory-bound problem (~335 MB gather traffic -> ~15us @ 23.3 TB/s).
// Kernel 1: tiny 67x67 Gauss-Jordan solve (constant LHS, 32 RHS cols).
// Kernel 2: per-wave 16-pixel tile; loc = PhiAug(16x68) @ W(68x32) via
//           v_wmma_f32_16x16x4_f32 chains, loc tile staged in LDS, then
//           fused bilinear gather with float4 (b128) loads/stores.
// ---------------------------------------------------------------------------

typedef __attribute__((ext_vector_type(2))) float v2f;
typedef __attribute__((ext_vector_type(8))) float v8f;

#define NN 67   // system size (64 cps + 3 affine)
#define NC 32   // rhs columns: 16 batches x (y,x)
#define NA 99   // augmented width: 67 + 32
#define KP 68   // K padded to multiple of 4

__device__ __forceinline__ float tps_phi(float r2) {
    return 0.5f * r2 * __logf(fmaxf(r2, 1e-10f));
}

// ---------------------------------------------------------------------------
// Kernel 1: build LHS + RHS, Gauss-Jordan with partial pivoting.
// Writes W transposed: wsW[c*KP + r] = solution row r, rhs column c (c=2*b+k).
// ---------------------------------------------------------------------------
__global__ void __launch_bounds__(128)
tps_solve(const float* __restrict__ off, float* __restrict__ wsW) {
    __shared__ float M[NN][NA + 1];   // +1 pad vs bank conflicts
    __shared__ float fac[NN];
    __shared__ int   piv;
    const int tid = threadIdx.x;

    // Build augmented matrix [lhs | rhs].
    for (int idx = tid; idx < NN * NA; idx += blockDim.x) {
        int i = idx / NA, j = idx % NA;
        float v;
        if (j < NN) {
            if (i < 64 && j < 64) {
                float dy = (float)(i >> 3) * (1.0f / 7.0f) - (float)(j >> 3) * (1.0f / 7.0f);
                float dx = (float)(i & 7)  * (1.0f / 7.0f) - (float)(j & 7)  * (1.0f / 7.0f);
                v = tps_phi(dy * dy + dx * dx);
            } else if (i < 64) {                 // B block: [cy, cx, 1]
                v = (j == 64) ? (float)(i >> 3) * (1.0f / 7.0f)
                  : (j == 65) ? (float)(i & 7)  * (1.0f / 7.0f) : 1.0f;
            } else if (j < 64) {                 // B^T block
                v = (i == 64) ? (float)(j >> 3) * (1.0f / 7.0f)
                  : (i == 65) ? (float)(j & 7)  * (1.0f / 7.0f) : 1.0f;
            } else {
                v = 0.0f;
            }
        } else {
            int c = j - NN, b = c >> 1, k = c & 1;
            if (i < 64) {
                float cpk = (k == 0) ? (float)(i >> 3) * (1.0f / 7.0f)
                                     : (float)(i & 7)  * (1.0f / 7.0f);
                v = cpk + off[b * 128 + i * 2 + k];
            } else {
                v = 0.0f;
            }
        }
        M[i][j] = v;
    }
    __syncthreads();

    // Gauss-Jordan with partial pivoting (TPS diagonal is zero -> mandatory).
    for (int ks = 0; ks < NN; ++ks) {
        if (tid == 0) {
            int p = ks; float best = fabsf(M[ks][ks]);
            for (int i = ks + 1; i < NN; ++i) {
                float a = fabsf(M[i][ks]);
                if (a > best) { best = a; p = i; }
            }
            piv = p;
        }
        __syncthreads();
        int p = piv;
        if (p != ks) {
            for (int j = tid; j < NA; j += blockDim.x) {
                float t = M[ks][j]; M[ks][j] = M[p][j]; M[p][j] = t;
            }
        }
        __syncthreads();
        float inv = 1.0f / M[ks][ks];
        for (int j = tid; j < NA; j += blockDim.x) M[ks][j] *= inv;
        __syncthreads();
        for (int i = tid; i < NN; i += blockDim.x)
            fac[i] = (i == ks) ? 0.0f : M[i][ks];
        __syncthreads();
        for (int idx = tid; idx < NN * NA; idx += blockDim.x) {
            int i = idx / NA, j = idx % NA;
            if (i != ks) M[i][j] -= fac[i] * M[ks][j];
        }
        __syncthreads();
    }

    // Emit W transposed with K padded to 68 (pad row = 0).
    for (int idx = tid; idx < NC * KP; idx += blockDim.x) {
        int c = idx / KP, r = idx % KP;
        wsW[c * KP + r] = (r < NN) ? M[r][NN + c] : 0.0f;
    }
}

// PhiAug[m][j]: j<64 -> TPS kernel vs control point j, 64->qy, 65->qx, 66->1, 67->0 (pad)
__device__ __forceinline__ float phi_aug(float qy, float qx, int j) {
    if (j < 64) {
        float dy = qy - (float)(j >> 3) * (1.0f / 7.0f);
        float dx = qx - (float)(j & 7)  * (1.0f / 7.0f);
        return tps_phi(dy * dy + dx * dx);
    }
    if (j == 64) return qy;
    if (j == 65) return qx;
    if (j == 66) return 1.0f;
    return 0.0f;
}

// ---------------------------------------------------------------------------
// Kernel 2: one wave per 16-pixel tile.
//   loc(16x32) = PhiAug(16x68) @ W(68x32) via 2 N-tiles x 17 K-steps of WMMA.
//   Then fused bilinear gather: 16 batches x 16 px x 4 float4 channel groups.
// ---------------------------------------------------------------------------
__global__ void __launch_bounds__(32)
tps_warp(const float* __restrict__ vol, const float* __restrict__ wsW,
         float* __restrict__ out) {
    __shared__ float locS[16][32];

    const int lane = threadIdx.x;
    const int half = lane >> 4;         // 0: K+0/K+1 lanes, 1: K+2/K+3 lanes
    const int lm   = lane & 15;
    const int m0   = blockIdx.x * 16;   // first pixel of tile
    const int m    = m0 + lm;
    const float qy = (float)(m >> 8) * (1.0f / 255.0f);
    const float qx = (float)(m & 255) * (1.0f / 255.0f);

    // A fragments for all 17 K-steps, computed once, reused for both N tiles.
    // 32-bit A 16x4 layout: lanes 0-15: {K0,K1}, lanes 16-31: {K2,K3}, M = lane%16.
    v2f afrag[17];
#pragma unroll
    for (int s = 0; s < 17; ++s) {
        int k0 = s * 4 + half * 2;
        v2f a;
        a.x = phi_aug(qy, qx, k0);
        a.y = phi_aug(qy, qx, k0 + 1);
        afrag[s] = a;
    }

    // Two 16-column N tiles of loc.
#pragma unroll
    for (int nt = 0; nt < 2; ++nt) {
        const int col = nt * 16 + lm;            // loc column (= 2*b + k)
        v8f c = {};
#pragma unroll
        for (int s = 0; s < 17; ++s) {
            int k0 = s * 4 + half * 2;
            // B 4x16 layout mirrors A: N = lane%16, lanes 0-15:{K0,K1}, 16-31:{K2,K3}.
            v2f b = *(const v2f*)(wsW + col * KP + k0);
            c = __builtin_amdgcn_wmma_f32_16x16x4_f32(
                    false, afrag[s], false, b, (short)0, c, false, false);
        }
        // D layout: VGPR v, lanes 0-15 -> row v, lanes 16-31 -> row v+8; col = lane%16.
#pragma unroll
        for (int v = 0; v < 8; ++v)
            locS[v + half * 8][nt * 16 + lm] = c[v];
    }
    __syncthreads();

    // Fused bilinear gather: per batch, 64 (px, channel-quad) items over 32 lanes.
    const int bstride = 256 * 256 * 16;
    for (int b = 0; b < 16; ++b) {
#pragma unroll
        for (int it = 0; it < 2; ++it) {
            int t  = lane + it * 32;   // 0..63
            int px = t >> 2;           // 0..15 pixel in tile
            int cg = t & 3;            // channel quad 0..3

            float yF = locS[px][2 * b]     * 255.0f;
            float xF = locS[px][2 * b + 1] * 255.0f;
            float y0f = floorf(yF), x0f = floorf(xF);
            float wy = yF - y0f,    wx = xF - x0f;
            int y0 = (int)y0f, x0 = (int)x0f;
            int y1 = y0 + 1,   x1 = x0 + 1;

            float vy0 = (y0 >= 0 && y0 < 256) ? 1.0f : 0.0f;
            float vy1 = (y1 >= 0 && y1 < 256) ? 1.0f : 0.0f;
            float vx0 = (x0 >= 0 && x0 < 256) ? 1.0f : 0.0f;
            float vx1 = (x1 >= 0 && x1 < 256) ? 1.0f : 0.0f;
            int y0c = min(max(y0, 0), 255), y1c = min(max(y1, 0), 255);
            int x0c = min(max(x0, 0), 255), x1c = min(max(x1, 0), 255);

            const float* base = vol + b * bstride + cg * 4;
            float4 g00 = *(const float4*)(base + (y0c * 256 + x0c) * 16);
            float4 g01 = *(const float4*)(base + (y0c * 256 + x1c) * 16);
            float4 g10 = *(const float4*)(base + (y1c * 256 + x0c) * 16);
            float4 g11 = *(const float4*)(base + (y1c * 256 + x1c) * 16);

            float w00 = (1.0f - wy) * (1.0f - wx) * vy0 * vx0;
            float w01 = (1.0f - wy) * wx          * vy0 * vx1;
            float w10 = wy          * (1.0f - wx) * vy1 * vx0;
            float w11 = wy          * wx          * vy1 * vx1;

            float4 o;
            o.x = g00.x * w00 + g01.x * w01 + g10.x * w10 + g11.x * w11;
            o.y = g00.y * w00 + g01.y * w01 + g10.y * w10 + g11.y * w11;
            o.z = g00.z * w00 + g01.z * w01 + g10.z * w10 + g11.z * w11;
            o.w = g00.w * w00 + g01.w * w01 + g10.w * w10 + g11.w * w11;

            int mm = m0 + px;
            *(float4*)(out + b * bstride + mm * 16 + cg * 4) = o;
        }
    }
}

extern "C" void kernel_launch(void* const* d_in, const int* in_sizes, int n_in,
                              void* d_out, int out_size, void* d_ws, size_t ws_size,
                              hipStream_t stream) {
    const float* vol = (const float*)d_in[0];   // (16,256,256,16) f32
    const float* off = (const float*)d_in[1];   // (16,64,2) f32
    float* wsW = (float*)d_ws;                  // 32*68 f32 = 8704 B
    float* out = (float*)d_out;                 // (16,256,256,16) f32

    tps_solve<<<1, 128, 0, stream>>>(off, wsW);
    tps_warp<<<4096, 32, 0, stream>>>(vol, wsW, out);
}